// MorphologicalDilation_62861141344686
// MI455X (gfx1250) — compile-verified
//
#include <hip/hip_runtime.h>

// Morphological dilation (tropical 3x3 conv), MI455X / gfx1250.
//
// out[b,y,x,f] = max_k ( x[b, y+ky, x+kx] + w[k, f] ),  VALID, stride 1
//   x: (16,256,256,1) f32, w: (1,1,1,9,32) f32, out: (16,254,254,32) f32
//
// Roofline: 132 MB of output writes dominate (input = 4 MB, lives in L2).
// Floor at 23.3 TB/s ~ 5.8 us; VALU cost (~13 lane-ops/pixel) is comparable,
// so minimize non-VALU issue slots:
//  * lane == filter (wave32 == 32 filters): weights in 9 VGPRs/lane,
//    perfectly coalesced 128B NT stores per wave per output pixel.
//  * input tile (+halo) staged via gfx1250 async global->LDS B128 loads
//    (ASYNCcnt / s_wait_asynccnt), row stride padded to 68 for alignment.
//  * float4 sliding window: 3 ds_load_b128 + 4 coalesced stores per
//    4 pixels; 4 independent pixel chains per iteration for dual-issue.

#define KH 3
#define KW 3
#define FILTERS 32
#define BATCH 16
#define H_IN 256
#define W_IN 256
#define HO (H_IN - KH + 1)   // 254
#define WO (W_IN - KW + 1)   // 254

#define TILE_H 8
#define TILE_W 64
#define IN_TH (TILE_H + KH - 1)      // 10
#define IN_TW 68                     // 66 needed, padded to mult. of 4
#define TILE_ELEMS (IN_TH * IN_TW)   // 680
#define ST_GROUPS (IN_TH * (IN_TW / 4))  // 170 float4 staging groups

typedef int v4i __attribute__((ext_vector_type(4)));

#if __has_builtin(__builtin_amdgcn_global_load_async_to_lds_b32)
#define HAVE_ASYNC32 1
#endif
#if __has_builtin(__builtin_amdgcn_global_load_async_to_lds_b128)
#define HAVE_ASYNC128 1
#endif

__global__ __launch_bounds__(256) void
dilation3x3_tropical_kernel(const float* __restrict__ x,
                            const float* __restrict__ w,
                            float* __restrict__ out)
{
    __shared__ float tile[TILE_ELEMS];

    const int lane = threadIdx.x;          // filter index (wave32 == 32 filters)
    const int wrow = threadIdx.y;          // wave id == local output row
    const int tid  = wrow * 32 + lane;
    const int ox0  = blockIdx.x * TILE_W;
    const int oy0  = blockIdx.y * TILE_H;
    const int b    = blockIdx.z;

    // ---- stage IN_TH x IN_TW input tile into LDS (async, B128 groups) ----
    const float* xb = x + (size_t)b * (H_IN * W_IN);
    if (tid < ST_GROUPS) {
        int ey  = tid / (IN_TW / 4);
        int ex4 = (tid - ey * (IN_TW / 4)) * 4;
        int gy  = oy0 + ey; if (gy > H_IN - 1) gy = H_IN - 1;   // clamp halo
        const float* grow = xb + gy * W_IN;
        float* lp = &tile[ey * IN_TW + ex4];
#if defined(HAVE_ASYNC128)
        if (ox0 + ex4 + 3 < W_IN) {
            __builtin_amdgcn_global_load_async_to_lds_b128(
                (v4i*)(grow + ox0 + ex4), (v4i*)lp, 0, 0);
        } else
#endif
        {
#pragma unroll
            for (int t = 0; t < 4; ++t) {
                int gx = ox0 + ex4 + t; if (gx > W_IN - 1) gx = W_IN - 1;
#if defined(HAVE_ASYNC32)
                __builtin_amdgcn_global_load_async_to_lds_b32(
                    (int*)(grow + gx), (int*)(lp + t), 0, 0);
#else
                lp[t] = grow[gx];
#endif
            }
        }
    }
#if defined(HAVE_ASYNC32) || defined(HAVE_ASYNC128)
#if __has_builtin(__builtin_amdgcn_s_wait_asynccnt)
    __builtin_amdgcn_s_wait_asynccnt(0);
#else
    asm volatile("s_wait_asynccnt 0" ::: "memory");
#endif
#endif
    __syncthreads();

    // ---- per-lane weights: lane f holds w[k][f], k = 0..8 -----------------
    float wv[KH * KW];
#pragma unroll
    for (int k = 0; k < KH * KW; ++k) wv[k] = w[k * FILTERS + lane];

    const int oy = oy0 + wrow;
    if (oy < HO) {
        const int rem   = WO - ox0;
        const int ncols = rem < TILE_W ? rem : TILE_W;   // 64 or 62
        const int r0 = wrow * IN_TW;                     // 16B-aligned groups
        const int r1 = r0 + IN_TW;
        const int r2 = r1 + IN_TW;

        size_t obase = (((size_t)b * HO + oy) * WO + ox0) * FILTERS + lane;

        // float4 sliding window: cur = cols[4g..4g+3], nxt = cols[4g+4..4g+7]
        float4 cur0 = *(const float4*)&tile[r0];
        float4 cur1 = *(const float4*)&tile[r1];
        float4 cur2 = *(const float4*)&tile[r2];

        const int ngrp = ncols >> 2;                     // 16 or 15
        for (int g = 0; g < ngrp; ++g) {
            float4 nx0 = *(const float4*)&tile[r0 + 4 * g + 4];
            float4 nx1 = *(const float4*)&tile[r1 + 4 * g + 4];
            float4 nx2 = *(const float4*)&tile[r2 + 4 * g + 4];

            float q0[6] = {cur0.x, cur0.y, cur0.z, cur0.w, nx0.x, nx0.y};
            float q1[6] = {cur1.x, cur1.y, cur1.z, cur1.w, nx1.x, nx1.y};
            float q2[6] = {cur2.x, cur2.y, cur2.z, cur2.w, nx2.x, nx2.y};

#pragma unroll
            for (int u = 0; u < 4; ++u) {
                // fmaxf trees -> v_max3_num_f32; 4 independent chains
                float m0 = fmaxf(fmaxf(q0[u] + wv[0], q0[u + 1] + wv[1]), q0[u + 2] + wv[2]);
                float m1 = fmaxf(fmaxf(q1[u] + wv[3], q1[u + 1] + wv[4]), q1[u + 2] + wv[5]);
                float m2 = fmaxf(fmaxf(q2[u] + wv[6], q2[u + 1] + wv[7]), q2[u + 2] + wv[8]);
                float m  = fmaxf(fmaxf(m0, m1), m2);
                // streamed output, never re-read: non-temporal store
                __builtin_nontemporal_store(
                    m, &out[obase + (size_t)(4 * g + u) * FILTERS]);
            }
            cur0 = nx0; cur1 = nx1; cur2 = nx2;
        }

        // tail: 0 or 2 pixels, window fully inside cur (cols 4*ngrp..4*ngrp+3)
        const int c = ngrp << 2;
        if (c < ncols) {
            float m0 = fmaxf(fmaxf(cur0.x + wv[0], cur0.y + wv[1]), cur0.z + wv[2]);
            float m1 = fmaxf(fmaxf(cur1.x + wv[3], cur1.y + wv[4]), cur1.z + wv[5]);
            float m2 = fmaxf(fmaxf(cur2.x + wv[6], cur2.y + wv[7]), cur2.z + wv[8]);
            __builtin_nontemporal_store(fmaxf(fmaxf(m0, m1), m2),
                                        &out[obase + (size_t)c * FILTERS]);
        }
        if (c + 1 < ncols) {
            float m0 = fmaxf(fmaxf(cur0.y + wv[0], cur0.z + wv[1]), cur0.w + wv[2]);
            float m1 = fmaxf(fmaxf(cur1.y + wv[3], cur1.z + wv[4]), cur1.w + wv[5]);
            float m2 = fmaxf(fmaxf(cur2.y + wv[6], cur2.z + wv[7]), cur2.w + wv[8]);
            __builtin_nontemporal_store(fmaxf(fmaxf(m0, m1), m2),
                                        &out[obase + (size_t)(c + 1) * FILTERS]);
        }
    }
}

extern "C" void kernel_launch(void* const* d_in, const int* in_sizes, int n_in,
                              void* d_out, int out_size, void* d_ws, size_t ws_size,
                              hipStream_t stream)
{
    (void)in_sizes; (void)n_in; (void)out_size; (void)d_ws; (void)ws_size;
    const float* x = (const float*)d_in[0];   // (16,256,256,1)
    const float* w = (const float*)d_in[1];   // (1,1,1,9,32)
    float* out = (float*)d_out;               // (16,254,254,32)

    dim3 grid((WO + TILE_W - 1) / TILE_W,     // 4
              (HO + TILE_H - 1) / TILE_H,     // 32
              BATCH);                         // 16
    dim3 block(32, TILE_H);                   // 8 waves (wave32)
    hipLaunchKernelGGL(dilation3x3_tropical_kernel, grid, block, 0, stream,
                       x, w, out);
}